// PointNetFeaturePropagation_38843684225787
// MI455X (gfx1250) — compile-verified
//
#include <hip/hip_runtime.h>
#include <hip/hip_bf16.h>
#include <math.h>

// ---------------------------------------------------------------------------
// PointNet Feature Propagation for MI455X (gfx1250), fp32 end-to-end.
// GEMMs use V_WMMA_F32_16X16X4_F32 (wave32 matrix core, full f32 precision).
//
// Shapes: B=16, N=4096, S=1024, C1=128, C2=256, CIN=384, CO=128, M=B*N=65536
// ---------------------------------------------------------------------------

#define B_   16
#define N_   4096
#define S_   1024
#define C1_  128
#define C2_  256
#define CIN_ 384
#define CO_  128
#define M_   (B_ * N_)   // 65536 rows

typedef __attribute__((ext_vector_type(2))) float v2f;
typedef __attribute__((ext_vector_type(8))) float v8f;

__device__ __forceinline__ float gelu_exact(float x) {
    return 0.5f * x * (1.0f + erff(x * 0.70710678118654752440f));
}

// ---------------------------------------------------------------------------
// Kernel 1: per query point, 3-NN over S=1024 reference points (per batch),
// inverse-distance weights (matches reference formula exactly).
// ---------------------------------------------------------------------------
__global__ __launch_bounds__(256) void knn3_kernel(const float* __restrict__ xyz1,
                                                   const float* __restrict__ xyz2,
                                                   int* __restrict__ out_idx,
                                                   float* __restrict__ out_w) {
    __shared__ float s_p[S_ * 3];   // xyz2 for this batch (12 KB)
    __shared__ float s_n[S_];       // |xyz2|^2 (4 KB)

    const int g = blockIdx.x * 256 + threadIdx.x;   // block stays within one batch
    const int b = g >> 12;                          // g / 4096

    for (int i = threadIdx.x; i < S_ * 3; i += 256) s_p[i] = xyz2[b * S_ * 3 + i];
    __syncthreads();
    for (int s = threadIdx.x; s < S_; s += 256) {
        const float x = s_p[s * 3], y = s_p[s * 3 + 1], z = s_p[s * 3 + 2];
        s_n[s] = x * x + y * y + z * z;
    }
    __syncthreads();

    const float px = xyz1[g * 3], py = xyz1[g * 3 + 1], pz = xyz1[g * 3 + 2];
    const float s1 = px * px + py * py + pz * pz;

    float d0 = 3.0e38f, d1 = 3.0e38f, d2 = 3.0e38f;
    int   i0 = 0, i1 = 0, i2 = 0;
    for (int s = 0; s < S_; ++s) {
        const float dot = px * s_p[s * 3] + py * s_p[s * 3 + 1] + pz * s_p[s * 3 + 2];
        const float sq  = s1 + s_n[s] - 2.0f * dot;
        const float d   = sqrtf(fmaxf(sq, 1e-12f));
        if (d < d0)      { d2 = d1; i2 = i1; d1 = d0; i1 = i0; d0 = d; i0 = s; }
        else if (d < d1) { d2 = d1; i2 = i1; d1 = d;  i1 = s; }
        else if (d < d2) { d2 = d;  i2 = s; }
    }
    const float r0 = 1.0f / (d0 + 1e-8f);
    const float r1 = 1.0f / (d1 + 1e-8f);
    const float r2 = 1.0f / (d2 + 1e-8f);
    const float rs = 1.0f / (r0 + r1 + r2);
    out_idx[g * 3 + 0] = i0; out_idx[g * 3 + 1] = i1; out_idx[g * 3 + 2] = i2;
    out_w[g * 3 + 0] = r0 * rs; out_w[g * 3 + 1] = r1 * rs; out_w[g * 3 + 2] = r2 * rs;
}

// ---------------------------------------------------------------------------
// Kernel 2: X0[row] = concat(points1[row] (128), interp(points2) (256))
// ---------------------------------------------------------------------------
__global__ __launch_bounds__(128) void build_x0_kernel(const float* __restrict__ points1,
                                                       const float* __restrict__ points2,
                                                       const int* __restrict__ idx,
                                                       const float* __restrict__ wgt,
                                                       float* __restrict__ X0) {
    const int row = blockIdx.x;            // 0..65535
    const int b   = row >> 12;
    const int c   = threadIdx.x;

    X0[(size_t)row * CIN_ + c] = points1[(size_t)row * C1_ + c];

    const int   j0 = idx[row * 3 + 0], j1 = idx[row * 3 + 1], j2 = idx[row * 3 + 2];
    const float w0 = wgt[row * 3 + 0], w1 = wgt[row * 3 + 1], w2 = wgt[row * 3 + 2];
    const float* p2 = points2 + (size_t)b * S_ * C2_;
    #pragma unroll
    for (int cc = c; cc < C2_; cc += 128) {
        const float v = w0 * p2[(size_t)j0 * C2_ + cc]
                      + w1 * p2[(size_t)j1 * C2_ + cc]
                      + w2 * p2[(size_t)j2 * C2_ + cc];
        X0[(size_t)row * CIN_ + C1_ + cc] = v;
    }
}

// ---------------------------------------------------------------------------
// Kernel 3: Y(M x 128) = X(M x K) * W^T(K x 128) + bias via
// V_WMMA_F32_16X16X4_F32.  256 threads (8 waves), 128x128 output tile,
// double-buffered LDS K-slabs of 16.
//
// sB is PAIR-INTERLEAVED: sB[(k>>1)*256 + o*2 + (k&1)], so a B fragment
// (K=ka, ka+1 for a given output column) is one aligned ds_load_b64 that
// lands directly in an even VGPR pair (WMMA operand requirement -> no movs).
//
// ISA §7.12.2 32-bit layouts:
//   A 16x4 : lane l: v0=A[M=l&15][K=2*(l>>4)],   v1=A[M=l&15][K=2*(l>>4)+1]
//   B 4x16 : lane l: v0=B[K=2*(l>>4)][N=l&15],   v1=B[K=2*(l>>4)+1][N=l&15]
//   C/D 16x16: vgpr r, lane l: M = r + 8*(l>>4), N = l&15
// ---------------------------------------------------------------------------
__global__ __launch_bounds__(256) void gemm_bias_wmma_kernel(const float* __restrict__ X,
                                                             const float* __restrict__ W,
                                                             const float* __restrict__ bias,
                                                             float* __restrict__ Y,
                                                             int K) {
    __shared__ alignas(16) float sA[2][128 * 16];   // [buf][row*16 + k]       2x8 KB
    __shared__ alignas(16) float sB[2][16 * 128];   // [buf][(k>>1)*256+o*2+(k&1)] 2x8 KB

    const int tid  = threadIdx.x;
    const int wave = tid >> 5;
    const int lane = tid & 31;
    const int lo   = lane & 15;
    const int hi   = lane >> 4;
    const int m0   = blockIdx.x * 128;

    auto stage = [&](int kbs, int buf) {
        // A slab: 128 rows x 16 k (coalesced float4)
        #pragma unroll
        for (int i = tid; i < 512; i += 256) {
            const int r  = i >> 2;
            const int kc = (i & 3) << 2;
            const float4 v = *reinterpret_cast<const float4*>(&X[(size_t)(m0 + r) * K + kbs + kc]);
            *reinterpret_cast<float4*>(&sA[buf][r * 16 + kc]) = v;
        }
        // B slab: W[o][k] -> pair-interleaved transpose, two ds_store_b64 each
        #pragma unroll
        for (int i = tid; i < 512; i += 256) {
            const int o  = i >> 2;
            const int kc = (i & 3) << 2;
            const float4 v = *reinterpret_cast<const float4*>(&W[(size_t)o * K + kbs + kc]);
            *reinterpret_cast<float2*>(&sB[buf][((kc >> 1) + 0) * 256 + o * 2]) = make_float2(v.x, v.y);
            *reinterpret_cast<float2*>(&sB[buf][((kc >> 1) + 1) * 256 + o * 2]) = make_float2(v.z, v.w);
        }
        // warm L2 for the slab after this one (global_prefetch_b8)
        if (kbs + 16 < K) {
            __builtin_prefetch(&X[(size_t)(m0 + (tid >> 1)) * K + kbs + 16 + ((tid & 1) << 3)], 0, 3);
            __builtin_prefetch(&W[(size_t)(tid >> 1) * K + kbs + 16 + ((tid & 1) << 3)], 0, 3);
        }
    };

    v8f acc[8];
    #pragma unroll
    for (int t = 0; t < 8; ++t)
        #pragma unroll
        for (int r = 0; r < 8; ++r) acc[t][r] = 0.0f;

    stage(0, 0);

    for (int kb = 0; kb < K; kb += 16) {
        const int buf = (kb >> 4) & 1;
        __syncthreads();                         // staging of `buf` visible; prev compute done
        if (kb + 16 < K) stage(kb + 16, buf ^ 1);  // overlap next staging with WMMAs

        const int arow = (wave << 4) | lo;
        #pragma unroll
        for (int kk = 0; kk < 16; kk += 4) {
            const int ka = kk + 2 * hi;
            const v2f a = *reinterpret_cast<const v2f*>(&sA[buf][arow * 16 + ka]);
            const int brow = (ka >> 1) * 256;    // == (kk/2 + hi) * 256
            #pragma unroll
            for (int t = 0; t < 8; ++t) {
                const v2f bf = *reinterpret_cast<const v2f*>(&sB[buf][brow + (t * 16 + lo) * 2]);
                acc[t] = __builtin_amdgcn_wmma_f32_16x16x4_f32(
                    false, a, false, bf, (short)0, acc[t], false, false);
            }
        }
    }

    // epilogue: C/D layout -> global, add bias
    #pragma unroll
    for (int t = 0; t < 8; ++t) {
        const int o  = t * 16 + lo;
        const float bv = bias[o];
        #pragma unroll
        for (int r = 0; r < 8; ++r) {
            const int m = m0 + (wave << 4) + r + (hi << 3);
            Y[(size_t)m * CO_ + o] = acc[t][r] + bv;
        }
    }
}

// ---------------------------------------------------------------------------
// BatchNorm statistics: zero -> block-partial atomic reduce -> finalize
// ---------------------------------------------------------------------------
__global__ void zero_stats_kernel(float* __restrict__ ssum, float* __restrict__ ssq) {
    const int c = threadIdx.x;   // 128 threads
    ssum[c] = 0.0f;
    ssq[c]  = 0.0f;
}

__global__ __launch_bounds__(256) void reduce_stats_kernel(const float* __restrict__ Y,
                                                           float* __restrict__ ssum,
                                                           float* __restrict__ ssq) {
    __shared__ float ls[256];
    __shared__ float lq[256];
    const int tid  = threadIdx.x;
    const int c    = tid & 127;
    const int half = tid >> 7;                // 0 or 1
    const int r0   = blockIdx.x * 256;        // 256 rows per block
    float s = 0.0f, q = 0.0f;
    for (int r = r0 + half; r < r0 + 256; r += 2) {
        const float v = Y[(size_t)r * CO_ + c];
        s += v;
        q += v * v;
    }
    ls[tid] = s; lq[tid] = q;
    __syncthreads();
    if (tid < 128) {
        s = ls[tid] + ls[tid + 128];
        q = lq[tid] + lq[tid + 128];
        atomicAdd(&ssum[c], s);
        atomicAdd(&ssq[c], q);
    }
}

__global__ void finalize_stats_kernel(const float* __restrict__ ssum,
                                      const float* __restrict__ ssq,
                                      float* __restrict__ mean,
                                      float* __restrict__ rstd,
                                      float invM) {
    const int c = threadIdx.x;   // 128 threads
    const float m = ssum[c] * invM;
    const float v = ssq[c] * invM - m * m;
    mean[c] = m;
    rstd[c] = rsqrtf(v + 1e-5f);
}

// ---------------------------------------------------------------------------
// Apply BN (+ optional residual) + optional exact GELU, float4-vectorized.
// ---------------------------------------------------------------------------
__global__ __launch_bounds__(256) void bn_act_kernel(const float* __restrict__ Y,
                                                     const float* __restrict__ mean,
                                                     const float* __restrict__ rstd,
                                                     const float* __restrict__ g,
                                                     const float* __restrict__ bt,
                                                     const float* __restrict__ resid,
                                                     float* __restrict__ out,
                                                     int do_gelu) {
    const int i4 = blockIdx.x * 256 + threadIdx.x;          // float4 index; grid covers M*CO/4
    const int cb = (i4 & ((CO_ >> 2) - 1)) << 2;            // channel base of this float4
    float4 y = reinterpret_cast<const float4*>(Y)[i4];
    float v[4] = {y.x, y.y, y.z, y.w};
    #pragma unroll
    for (int j = 0; j < 4; ++j) {
        const int c = cb + j;
        v[j] = (v[j] - mean[c]) * rstd[c] * g[c] + bt[c];
    }
    if (resid) {
        const float4 r = reinterpret_cast<const float4*>(resid)[i4];
        v[0] += r.x; v[1] += r.y; v[2] += r.z; v[3] += r.w;
    }
    if (do_gelu) {
        #pragma unroll
        for (int j = 0; j < 4; ++j) v[j] = gelu_exact(v[j]);
    }
    reinterpret_cast<float4*>(out)[i4] = make_float4(v[0], v[1], v[2], v[3]);
}

// ---------------------------------------------------------------------------
// Host-side orchestration
// ---------------------------------------------------------------------------
extern "C" void kernel_launch(void* const* d_in, const int* in_sizes, int n_in,
                              void* d_out, int out_size, void* d_ws, size_t ws_size,
                              hipStream_t stream) {
    const float* xyz1    = (const float*)d_in[0];
    const float* xyz2    = (const float*)d_in[1];
    const float* points1 = (const float*)d_in[2];
    const float* points2 = (const float*)d_in[3];
    const float* fuse_w  = (const float*)d_in[4];
    const float* fuse_b  = (const float*)d_in[5];
    const float* fuse_g  = (const float*)d_in[6];
    const float* fuse_bt = (const float*)d_in[7];
    const float* w1      = (const float*)d_in[8];
    const float* b1      = (const float*)d_in[9];
    const float* g1      = (const float*)d_in[10];
    const float* bt1     = (const float*)d_in[11];
    const float* w2      = (const float*)d_in[12];
    const float* b2      = (const float*)d_in[13];
    const float* g2      = (const float*)d_in[14];
    const float* bt2     = (const float*)d_in[15];

    // workspace carve-up (floats)
    float* X0   = (float*)d_ws;                       // M x 384
    float* X1   = X0 + (size_t)M_ * CIN_;             // M x 128 (post fuse BN+GELU; residual src)
    float* Hbuf = X1 + (size_t)M_ * CO_;              // M x 128
    int*   gidx = (int*)(Hbuf + (size_t)M_ * CO_);    // M x 3
    float* gw   = (float*)(gidx + (size_t)M_ * 3);    // M x 3
    float* ssum = gw + (size_t)M_ * 3;                // 128
    float* ssq  = ssum + CO_;                         // 128
    float* smean= ssq + CO_;                          // 128
    float* srstd= smean + CO_;                        // 128

    float* Y = (float*)d_out;   // GEMM scratch aliases output (safe: elementwise same-index)

    const float invM = 1.0f / (float)M_;
    const int elem4_blocks = (M_ * CO_) / 4 / 256;    // 8192

    // 1) 3-NN + weights
    knn3_kernel<<<M_ / 256, 256, 0, stream>>>(xyz1, xyz2, gidx, gw);
    // 2) interpolate + concat
    build_x0_kernel<<<M_, 128, 0, stream>>>(points1, points2, gidx, gw, X0);

    // 3) fuse layer: GEMM(384->128) + BN + GELU -> X1
    gemm_bias_wmma_kernel<<<M_ / 128, 256, 0, stream>>>(X0, fuse_w, fuse_b, Y, CIN_);
    zero_stats_kernel<<<1, 128, 0, stream>>>(ssum, ssq);
    reduce_stats_kernel<<<M_ / 256, 256, 0, stream>>>(Y, ssum, ssq);
    finalize_stats_kernel<<<1, 128, 0, stream>>>(ssum, ssq, smean, srstd, invM);
    bn_act_kernel<<<elem4_blocks, 256, 0, stream>>>(Y, smean, srstd, fuse_g, fuse_bt,
                                                    nullptr, X1, 1);

    // 4) layer 1: GEMM(128->128) + BN + GELU -> Hbuf
    gemm_bias_wmma_kernel<<<M_ / 128, 256, 0, stream>>>(X1, w1, b1, Y, CO_);
    zero_stats_kernel<<<1, 128, 0, stream>>>(ssum, ssq);
    reduce_stats_kernel<<<M_ / 256, 256, 0, stream>>>(Y, ssum, ssq);
    finalize_stats_kernel<<<1, 128, 0, stream>>>(ssum, ssq, smean, srstd, invM);
    bn_act_kernel<<<elem4_blocks, 256, 0, stream>>>(Y, smean, srstd, g1, bt1,
                                                    nullptr, Hbuf, 1);

    // 5) layer 2: GEMM(128->128) + BN, residual add X1, GELU -> d_out
    gemm_bias_wmma_kernel<<<M_ / 128, 256, 0, stream>>>(Hbuf, w2, b2, Y, CO_);
    zero_stats_kernel<<<1, 128, 0, stream>>>(ssum, ssq);
    reduce_stats_kernel<<<M_ / 256, 256, 0, stream>>>(Y, ssum, ssq);
    finalize_stats_kernel<<<1, 128, 0, stream>>>(ssum, ssq, smean, srstd, invM);
    bn_act_kernel<<<elem4_blocks, 256, 0, stream>>>(Y, smean, srstd, g2, bt2,
                                                    X1, (float*)d_out, 1);
}